// MultipleChoiceLoss_64931315581274
// MI455X (gfx1250) — compile-verified
//
#include <hip/hip_runtime.h>
#include <hip/hip_bf16.h>

typedef __attribute__((ext_vector_type(2))) float v2f;
typedef __attribute__((ext_vector_type(8))) float v8f;

#define BLOCKS   1024
#define THREADS  256
#define NCLASS   5

// Pass 1: grid-stride hinge partial sums, one float partial per block -> d_ws.
__global__ void __launch_bounds__(THREADS)
hinge_partial_kernel(const float* __restrict__ score,
                     const int*   __restrict__ tgt_lo,   // low dwords of int64 targets
                     float* __restrict__ partials,
                     int n)
{
    int idx    = blockIdx.x * THREADS + threadIdx.x;
    int stride = gridDim.x * THREADS;

    float acc = 0.0f;
    for (int i = idx; i < n; i += stride) {
        const float* r = score + i * NCLASS;
        float s0 = r[0], s1 = r[1], s2 = r[2], s3 = r[3], s4 = r[4];
        int tg = tgt_lo[2 * i];  // little-endian low word of int64, value in [0,4]
        float gold = (tg == 0) ? s0 :
                     (tg == 1) ? s1 :
                     (tg == 2) ? s2 :
                     (tg == 3) ? s3 : s4;
        float m = 1.0f - gold;
        // sum over all c of max(0, 1 + s[c] - gold); the c==tg term is exactly 1.0
        acc += fmaxf(0.0f, s0 + m) + fmaxf(0.0f, s1 + m) + fmaxf(0.0f, s2 + m) +
               fmaxf(0.0f, s3 + m) + fmaxf(0.0f, s4 + m) - 1.0f;
    }

    // wave32 reduction
    for (int off = 16; off > 0; off >>= 1)
        acc += __shfl_down(acc, off, 32);

    __shared__ float wsum[THREADS / 32];
    int lane = threadIdx.x & 31;
    int wid  = threadIdx.x >> 5;
    if (lane == 0) wsum[wid] = acc;
    __syncthreads();

    if (threadIdx.x == 0) {
        float b = 0.0f;
        #pragma unroll
        for (int w = 0; w < THREADS / 32; ++w) b += wsum[w];
        partials[blockIdx.x] = b;
    }
}

// Pass 2: single wave32. Sum `nb` partials (nb % 64 == 0) through the matrix
// pipe: D = A(ones 16x4) x B(4x16 of data) + C. Since A is all ones, every row
// of D holds the per-column sums of B, so sum(row 0 of C) == sum of all 64
// elements fed into B each iteration — layout-agnostic.
__global__ void __launch_bounds__(32)
finalize_wmma_kernel(const float* __restrict__ partials,
                     float* __restrict__ out,
                     float inv_cnt, int nb)
{
    int lane = threadIdx.x;           // 0..31, EXEC all ones (WMMA requirement)
    v2f a; a.x = 1.0f; a.y = 1.0f;    // ones A-matrix
    v8f c = {0.f, 0.f, 0.f, 0.f, 0.f, 0.f, 0.f, 0.f};

    for (int t = 0; t < nb; t += 64) {
        v2f b;
        b.x = partials[t + lane];
        b.y = partials[t + 32 + lane];
        // 8 args: (neg_a, A, neg_b, B, c_mod, C, reuse_a, reuse_b)
        c = __builtin_amdgcn_wmma_f32_16x16x4_f32(
                false, a, false, b, (short)0, c, false, false);
    }

    // c[0]: lanes 0-15 = D[0][n], lanes 16-31 = D[8][n] (identical rows).
    // Full 32-lane butterfly gives 2x the total.
    float s = c[0];
    for (int m = 16; m > 0; m >>= 1)
        s += __shfl_xor(s, m, 32);

    if (lane == 0)
        out[0] = s * 0.5f * inv_cnt;
}

extern "C" void kernel_launch(void* const* d_in, const int* in_sizes, int n_in,
                              void* d_out, int out_size, void* d_ws, size_t ws_size,
                              hipStream_t stream)
{
    const float* score  = (const float*)d_in[0];
    const int*   tgt_lo = (const int*)d_in[1];   // int64 targets, read low dwords
    float*       out    = (float*)d_out;
    float*       parts  = (float*)d_ws;          // BLOCKS floats of scratch

    int n = in_sizes[0] / NCLASS;                // 4,000,000 rows
    float inv_cnt = 1.0f / (float)(n * (NCLASS - 1));

    hinge_partial_kernel<<<BLOCKS, THREADS, 0, stream>>>(score, tgt_lo, parts, n);
    finalize_wmma_kernel<<<1, 32, 0, stream>>>(parts, out, inv_cnt, BLOCKS);
}